// SoftECE_27779848471442
// MI455X (gfx1250) — compile-verified
//
#include <hip/hip_runtime.h>
#include <hip/hip_bf16.h>
#include <float.h>
#include <stdint.h>

// ---------------------------------------------------------------------------
// ECE over [N=524288, C=100] fp32 logits, labels int32.
// Memory bound: 210 MB single pass -> ~9us floor at 23.3 TB/s.
// Design: 16-row (6400B contiguous) tiles async-DMA'd global->LDS
// (GLOBAL_LOAD_ASYNC_TO_LDS_B128, ASYNCcnt), double buffered per wave;
// per-row exp-sum via V_WMMA_F32_16X16X4_F32 with A=ones (exact fp32 batched
// reduction of 16 rows); pred prob via one dynamic LDS read of row[label]
// (no per-element compare/select chain); 15-bin per-block LDS accum ->
// 15 global f32 atomics per block -> 1-thread finish kernel.
// Identity used: ece = sum_b | sum_{rows in b} (max_p - pred_p) | / N.
// ---------------------------------------------------------------------------

#define NUM_BINS 15
#define NCLASS 100
#define ROWS_PER_TILE 16
#define TILE_FLOATS (ROWS_PER_TILE * NCLASS)        // 1600 floats = 6400 B
#define TILE_B128 ((TILE_FLOATS * 4) / 16)          // 400 16-byte chunks
#define ASYNC_OPS_PER_TILE ((TILE_B128 + 31) / 32)  // 13 wave-level async ops
#define WAVES_PER_BLOCK 4
#define BLOCK_THREADS (WAVES_PER_BLOCK * 32)
#define GRID_BLOCKS 2048

typedef __attribute__((ext_vector_type(2))) float v2f;
typedef __attribute__((ext_vector_type(8))) float v8f;
typedef __attribute__((ext_vector_type(4))) int v4i;

#ifndef __has_builtin
#define __has_builtin(x) 0
#endif

#if __has_builtin(__builtin_amdgcn_wmma_f32_16x16x4_f32)
#define HAVE_WMMA_F32_K4 1
#else
#define HAVE_WMMA_F32_K4 0
#endif

#if __has_builtin(__builtin_amdgcn_global_load_async_to_lds_b128)
#define HAVE_ASYNC_BUILTIN 1
#else
#define HAVE_ASYNC_BUILTIN 0
#endif

// ---- async global -> LDS tile copy (16 rows, contiguous 6400B) -------------
__device__ __forceinline__ void issue_tile(const float* __restrict__ gsrc,
                                           float* ldsdst, int lane) {
  const char* g = (const char*)gsrc;
  uint32_t lbase = (uint32_t)(uintptr_t)ldsdst;  // low 32 bits = LDS byte addr
  for (int i = lane; i < TILE_B128; i += 32) {
#if HAVE_ASYNC_BUILTIN
    // Builtin signature (from clang diagnostic): (v4i AS1*, v4i AS3*, Ii, Ii)
    __builtin_amdgcn_global_load_async_to_lds_b128(
        (__attribute__((address_space(1))) v4i*)(uintptr_t)(g + (size_t)i * 16),
        (__attribute__((address_space(3))) v4i*)(uintptr_t)(lbase + (uint32_t)i * 16u),
        0, 0);
#else
    uint64_t ga = (uint64_t)(uintptr_t)(g + (size_t)i * 16);
    uint32_t la = lbase + (uint32_t)i * 16u;
    asm volatile("global_load_async_to_lds_b128 %0, %1, off"
                 :: "v"(la), "v"(ga) : "memory");
#endif
  }
}

__device__ __forceinline__ void wait_async_keep_next() {
#if __has_builtin(__builtin_amdgcn_s_wait_asynccnt)
  __builtin_amdgcn_s_wait_asynccnt(ASYNC_OPS_PER_TILE);
#else
  asm volatile("s_wait_asynccnt %0" :: "i"(ASYNC_OPS_PER_TILE) : "memory");
#endif
}

__device__ __forceinline__ void wait_async_all() {
#if __has_builtin(__builtin_amdgcn_s_wait_asynccnt)
  __builtin_amdgcn_s_wait_asynccnt(0);
#else
  asm volatile("s_wait_asynccnt 0" ::: "memory");
#endif
}

// ---- process one 16-row tile resident in LDS -------------------------------
__device__ __forceinline__ void process_tile(const float* __restrict__ tile,
                                             const int* __restrict__ labels,
                                             int row0, int lane,
                                             float* __restrict__ bins_s) {
  const int r = lane & 15;        // my row within the tile
  const int h = lane >> 4;        // half: cols {0,1} vs {2,3} mod 4
  const float* myrow = tile + r * NCLASS;

  // ---- pass 1: row max (each lane scans its 50 columns; conflict-free) ----
  float mloc = -FLT_MAX;
#pragma unroll
  for (int k0 = 0; k0 < NCLASS; k0 += 4) {
    float2 x = *(const float2*)(myrow + k0 + 2 * h);
    mloc = fmaxf(mloc, fmaxf(x.x, x.y));
  }
  const float m = fmaxf(mloc, __shfl_xor(mloc, 16, 32));

  // ---- pass 2: exp + batched row-sum via fp32 WMMA with A = ones ----------
#if HAVE_WMMA_F32_K4
  v8f acc = {0.f, 0.f, 0.f, 0.f, 0.f, 0.f, 0.f, 0.f};
  const v2f ones = {1.0f, 1.0f};
#else
  float sloc = 0.0f;
#endif
#pragma unroll
  for (int k0 = 0; k0 < NCLASS; k0 += 4) {
    float2 x = *(const float2*)(myrow + k0 + 2 * h);
    const float e0 = __expf(x.x - m);
    const float e1 = __expf(x.y - m);
#if HAVE_WMMA_F32_K4
    // D[m][n] += sum_k B[k][n] since A==1; lane n (=row) gets its row sum
    // replicated in every accumulator VGPR. Exact fp32 math.
    v2f b = {e0, e1};
    acc = __builtin_amdgcn_wmma_f32_16x16x4_f32(
        false, ones, false, b, (short)0, acc, false, false);
#else
    sloc += e0 + e1;
#endif
  }
#if HAVE_WMMA_F32_K4
  const float s = acc[0];                       // row (lane&15) sum, any VGPR
#else
  const float s = sloc + __shfl_xor(sloc, 16, 32);
#endif

  if (lane < 16) {                              // one finisher lane per row
    const int lab = labels[row0 + r];           // EXEC=half: 16 b32 loads
    const float xlab = myrow[lab];              // one dynamic LDS read
    const float pred = __expf(xlab - m);        // exp(x[label] - max)
    const float conf = 1.0f / s;                // max_prob = exp(m-m)/s
    const float diff = (1.0f - pred) * conf;    // max_prob - pred_prob
    int bin = (int)(conf * 15.0f);
    bin = bin < 0 ? 0 : (bin > NUM_BINS - 1 ? NUM_BINS - 1 : bin);
    atomicAdd(&bins_s[bin], diff);              // ds_add_f32
  }
}

// ---- kernels ---------------------------------------------------------------
__global__ void ece_zero_ws(float* ws) {
  if (threadIdx.x < 16) ws[threadIdx.x] = 0.0f;
}

__global__ void __launch_bounds__(BLOCK_THREADS)
ece_main(const float* __restrict__ logits, const int* __restrict__ labels,
         float* __restrict__ ws, int nrows) {
  __shared__ float tiles[WAVES_PER_BLOCK * 2 * TILE_FLOATS];  // 51.2 KB
  __shared__ float bins_s[NUM_BINS];

  const int tid = threadIdx.x;
  const int lane = tid & 31;
  const int wave = tid >> 5;
  if (tid < NUM_BINS) bins_s[tid] = 0.0f;
  __syncthreads();

  const int ntiles = nrows / ROWS_PER_TILE;
  const int nwaves = (int)gridDim.x * WAVES_PER_BLOCK;
  int t = (int)blockIdx.x * WAVES_PER_BLOCK + wave;

  float* buf0 = tiles + wave * 2 * TILE_FLOATS;
  float* buf1 = buf0 + TILE_FLOATS;

  if (t < ntiles)
    issue_tile(logits + (size_t)t * ROWS_PER_TILE * NCLASS, buf0, lane);

  int p = 0;
  for (; t < ntiles; t += nwaves) {
    const int tn = t + nwaves;
    float* cur = p ? buf1 : buf0;
    float* nxt = p ? buf0 : buf1;
    if (tn < ntiles) {
      issue_tile(logits + (size_t)tn * ROWS_PER_TILE * NCLASS, nxt, lane);
      wait_async_keep_next();   // cnt<=13: current tile done, next in flight
    } else {
      wait_async_all();
    }
    process_tile(cur, labels, t * ROWS_PER_TILE, lane, bins_s);
    p ^= 1;
  }

  __syncthreads();
  if (tid < NUM_BINS) atomicAdd(&ws[tid], bins_s[tid]);  // 15 atomics/block
}

__global__ void ece_finish(const float* __restrict__ ws, float* __restrict__ out,
                           int nrows) {
  if (threadIdx.x == 0) {
    float s = 0.0f;
    for (int b = 0; b < NUM_BINS; ++b) s += fabsf(ws[b]);
    out[0] = s / (float)nrows;
  }
}

// ---- launcher --------------------------------------------------------------
extern "C" void kernel_launch(void* const* d_in, const int* in_sizes, int n_in,
                              void* d_out, int out_size, void* d_ws, size_t ws_size,
                              hipStream_t stream) {
  const float* logits = (const float*)d_in[0];
  const int* labels = (const int*)d_in[1];
  const int nrows = in_sizes[1];  // N (labels count); logits = N*100
  float* ws = (float*)d_ws;       // 15 bin accumulators
  float* out = (float*)d_out;

  ece_zero_ws<<<1, 32, 0, stream>>>(ws);
  ece_main<<<GRID_BLOCKS, BLOCK_THREADS, 0, stream>>>(logits, labels, ws, nrows);
  ece_finish<<<1, 32, 0, stream>>>(ws, out, nrows);
}